// Encoder_Model_58969900974821
// MI455X (gfx1250) — compile-verified
//
#include <hip/hip_runtime.h>
#include <hip/hip_bf16.h>
#include <stdint.h>
#include <stddef.h>

#define N_NODES   100000
#define HIDDEN    128
#define OSTRIDE   384   // 3 * 128, concat layout

// --- CDNA5 async global->LDS gather path (probe via __has_builtin) ---------
#if defined(__has_builtin)
#  if __has_builtin(__builtin_amdgcn_global_load_async_to_lds_b128) && \
      __has_builtin(__builtin_amdgcn_s_wait_asynccnt)
#    define USE_ASYNC_LDS 1
#  endif
#endif

#define AS1 __attribute__((address_space(1)))
#define AS3 __attribute__((address_space(3)))
typedef int v4i __attribute__((ext_vector_type(4)));

// ---------------------------------------------------------------------------
// Pass 1: per-row degree (segment_sum of ones)
__global__ __launch_bounds__(256) void deg_count_kernel(
    const int* __restrict__ rows, unsigned* __restrict__ deg, int E) {
  int i = blockIdx.x * blockDim.x + threadIdx.x;
  if (i < E) atomicAdd(&deg[rows[i]], 1u);
}

// Pass 2: count -> softmax-of-ones weight = 1 / max(deg, 1)
__global__ __launch_bounds__(256) void deg_recip_kernel(
    const unsigned* __restrict__ deg, float* __restrict__ wrec, int n) {
  int i = blockIdx.x * blockDim.x + threadIdx.x;
  if (i < n) {
    unsigned d = deg[i];
    wrec[i] = 1.0f / (float)(d ? d : 1u);
  }
}

// Pass 3: gather emb[col,:], scale by w[row], atomically scatter-add into
// out[row, colOff : colOff+128]. One wave32 per edge; lane l owns floats
// [4l, 4l+4). Grid-stride chain per wave with double-buffered async LDS
// staging so the next edge's 512B row is in flight while we drain this one.
__global__ __launch_bounds__(256) void scatter_avg_kernel(
    const int* __restrict__ rows, const int* __restrict__ cols,
    const float* __restrict__ emb, const float* __restrict__ wrec,
    float* __restrict__ out, int E, int colOff) {
  const int lane = threadIdx.x & 31;
  const int wib  = threadIdx.x >> 5;                     // wave in block (0..7)
  const int gw   = blockIdx.x * (blockDim.x >> 5) + wib; // global wave id
  const int nw   = gridDim.x * (blockDim.x >> 5);        // total waves

  if (gw >= E) return;  // wave-uniform exit, EXEC stays all-ones

#ifdef USE_ASYNC_LDS
  __shared__ float buf[8][2][HIDDEN];   // 8 waves x 2 stages x 512B = 8KB
  int par = 0;
  {
    const float* src = emb + (size_t)cols[gw] * HIDDEN + lane * 4;
    __builtin_amdgcn_global_load_async_to_lds_b128(
        (AS1 v4i*)src, (AS3 v4i*)&buf[wib][0][lane * 4], 0, 0);
  }
  for (int e = gw; e < E; e += nw) {
    const int nexte = e + nw;
    if (nexte < E) {
      const float* src = emb + (size_t)cols[nexte] * HIDDEN + lane * 4;
      __builtin_amdgcn_global_load_async_to_lds_b128(
          (AS1 v4i*)src, (AS3 v4i*)&buf[wib][par ^ 1][lane * 4], 0, 0);
      __builtin_amdgcn_s_wait_asynccnt(1);  // current stage landed in LDS
    } else {
      __builtin_amdgcn_s_wait_asynccnt(0);
    }
    const int   row = rows[e];
    const float w   = wrec[row];
    float4 v = *(const float4*)&buf[wib][par][lane * 4];  // ds_load_b128
    float* op = out + (size_t)row * OSTRIDE + colOff + lane * 4;
    unsafeAtomicAdd(op + 0, v.x * w);   // global_atomic_add_f32, no-return
    unsafeAtomicAdd(op + 1, v.y * w);
    unsafeAtomicAdd(op + 2, v.z * w);
    unsafeAtomicAdd(op + 3, v.w * w);
    par ^= 1;
  }
#else
  for (int e = gw; e < E; e += nw) {
    const int   row = rows[e];
    const int   col = cols[e];
    const float w   = wrec[row];
    float4 v = *(const float4*)(emb + (size_t)col * HIDDEN + lane * 4);
    float* op = out + (size_t)row * OSTRIDE + colOff + lane * 4;
    unsafeAtomicAdd(op + 0, v.x * w);
    unsafeAtomicAdd(op + 1, v.y * w);
    unsafeAtomicAdd(op + 2, v.z * w);
    unsafeAtomicAdd(op + 3, v.w * w);
  }
#endif
}

// ---------------------------------------------------------------------------
static inline int cdiv(int a, int b) { return (a + b - 1) / b; }

extern "C" void kernel_launch(void* const* d_in, const int* in_sizes, int n_in,
                              void* d_out, int out_size, void* d_ws, size_t ws_size,
                              hipStream_t stream) {
  const int*   ent_adj = (const int*)d_in[0];
  const int*   rel_adj = (const int*)d_in[1];
  const int*   att_adj = (const int*)d_in[2];
  const float* ent_emb = (const float*)d_in[3];
  const float* rel_emb = (const float*)d_in[4];
  const float* att_emb = (const float*)d_in[5];
  float* out = (float*)d_out;

  const int E_ent = in_sizes[0] / 2;
  const int E_rel = in_sizes[1] / 2;
  const int E_att = in_sizes[2] / 2;

  // workspace: [0, 3N) u32 degree counters | [3N, 6N) f32 reciprocal weights
  unsigned* deg  = (unsigned*)d_ws;
  float*    wrec = (float*)((char*)d_ws + (size_t)3 * N_NODES * sizeof(unsigned));

  (void)hipMemsetAsync(deg, 0, (size_t)3 * N_NODES * sizeof(unsigned), stream);
  (void)hipMemsetAsync(out, 0, (size_t)out_size * sizeof(float), stream);

  deg_count_kernel<<<cdiv(E_ent, 256), 256, 0, stream>>>(ent_adj, deg + 0 * N_NODES, E_ent);
  deg_count_kernel<<<cdiv(E_rel, 256), 256, 0, stream>>>(rel_adj, deg + 1 * N_NODES, E_rel);
  deg_count_kernel<<<cdiv(E_att, 256), 256, 0, stream>>>(att_adj, deg + 2 * N_NODES, E_att);

  deg_recip_kernel<<<cdiv(3 * N_NODES, 256), 256, 0, stream>>>(deg, wrec, 3 * N_NODES);

  const int SBLK = 4096;  // 32768 waves -> chains of ~25-50 edges per wave
  scatter_avg_kernel<<<SBLK, 256, 0, stream>>>(
      ent_adj, ent_adj + E_ent, ent_emb, wrec + 0 * N_NODES, out, E_ent, 0);
  scatter_avg_kernel<<<SBLK, 256, 0, stream>>>(
      rel_adj, rel_adj + E_rel, rel_emb, wrec + 1 * N_NODES, out, E_rel, 128);
  scatter_avg_kernel<<<SBLK, 256, 0, stream>>>(
      att_adj, att_adj + E_att, att_emb, wrec + 2 * N_NODES, out, E_att, 256);
}